// GraphOfGraphsModel_90099823935523
// MI455X (gfx1250) — compile-verified
//
#include <hip/hip_runtime.h>

// ---------------------------------------------------------------------------
// GraphConv x2 + mean-pool + FC for MI455X (gfx1250, wave32).
// Dense GEMMs use V_WMMA_F32_16X16X4_F32 (full fp32 precision).
// Edge aggregation uses coalesced float4 gathers + global_atomic_add_f32.
// ---------------------------------------------------------------------------

typedef __attribute__((ext_vector_type(2))) float v2f;
typedef __attribute__((ext_vector_type(8))) float v8f;

// LDS layout for transposed, k-paired weights.
// pair index p in [0,64): holds (W^T[2p][n], W^T[2p+1][n]) = (W[n][2p], W[n][2p+1])
// Additive swizzle spreads banks for the transposing stores.
#define WIDX(p, n) ((p) * 128 + (((n) + (p)) & 127))

__global__ void zero_kernel(float* __restrict__ p, int n /* multiple of 4 */) {
    int i4 = (blockIdx.x * blockDim.x + threadIdx.x) * 4;
    if (i4 < n) *(float4*)(p + i4) = make_float4(0.f, 0.f, 0.f, 0.f);
}

// One edge per wave: 32 lanes x float4 = 128 floats.
__global__ void scatter_add(const float* __restrict__ x, const int* __restrict__ src,
                            const int* __restrict__ dst, float* __restrict__ agg, int E) {
    int tid  = blockIdx.x * 256 + threadIdx.x;
    int e    = tid >> 5;
    if (e >= E) return;
    int l    = tid & 31;
    int s    = src[e];
    int d    = dst[e];
    float4 v = *(const float4*)(x + (long long)s * 128 + l * 4);
    float* a = agg + (long long)d * 128 + l * 4;
    atomicAdd(a + 0, v.x);
    atomicAdd(a + 1, v.y);
    atomicAdd(a + 2, v.z);
    atomicAdd(a + 3, v.w);
}

// out[m][:] = relu( A1[m]·Wrel^T + A2[m]·Wroot^T + bias ),  m in 16-row stripes/wave.
// Block = 256 threads = 8 waves -> 128 rows per block. D = H = 128 fixed.
__global__ __launch_bounds__(256) void gemm2_bias_relu(
    const float* __restrict__ A1, const float* __restrict__ A2,
    const float* __restrict__ Wrel, const float* __restrict__ Wroot,
    const float* __restrict__ bias, float* __restrict__ out, int nrows) {
    __shared__ float2 ldsW[8192];  // 64 KB: one 128x128 weight, transposed + k-paired

    const int tid      = threadIdx.x;
    const int lane     = tid & 31;
    const int wave     = tid >> 5;
    const int half     = lane >> 4;   // 0: k+0/k+1 lanes, 1: k+2/k+3 lanes
    const int l16      = lane & 15;
    const int row_base = blockIdx.x * 128 + wave * 16;
    const bool active  = (row_base + 16) <= nrows;  // wave-uniform: EXEC all-1s inside

    // Bias-initialized accumulators. C/D layout: VGPR j -> M = j + half*8, N = l16.
    v8f acc[8];
    for (int nb = 0; nb < 8; ++nb) {
        float bv = bias[nb * 16 + l16];
        for (int j = 0; j < 8; ++j) acc[nb][j] = bv;
    }

    const float* mats[2] = {A1, A2};
    const float* wts[2]  = {Wrel, Wroot};

    for (int phase = 0; phase < 2; ++phase) {
        const float* W = wts[phase];
        __syncthreads();  // previous-phase LDS reads done
        // Stage W^T into LDS: coalesced float4 row reads, paired transposed stores.
        for (int idx = tid; idx < 4096; idx += 256) {
            int r  = idx >> 5;          // weight row n (output feature)
            int c4 = (idx & 31) << 2;   // k
            float4 f = *(const float4*)(W + r * 128 + c4);
            int p0 = c4 >> 1;
            ldsW[WIDX(p0, r)]     = make_float2(f.x, f.y);
            ldsW[WIDX(p0 + 1, r)] = make_float2(f.z, f.w);
        }
        __syncthreads();

        if (active) {
            // A frag: lane holds A[row_base+l16][4kb + 2*half + {0,1}]
            const float* Abase = mats[phase] + (long long)(row_base + l16) * 128 + 2 * half;
            for (int kb = 0; kb < 32; ++kb) {
                float2 av = *(const float2*)(Abase + 4 * kb);
                v2f a;
                a[0] = av.x;
                a[1] = av.y;
                const int p = 2 * kb + half;
                for (int nb = 0; nb < 8; ++nb) {
                    float2 bw = ldsW[WIDX(p, nb * 16 + l16)];
                    v2f b;
                    b[0] = bw.x;
                    b[1] = bw.y;
                    acc[nb] = __builtin_amdgcn_wmma_f32_16x16x4_f32(
                        false, a, false, b, (short)0, acc[nb], false, false);
                }
            }
        }
    }

    if (active) {
        for (int nb = 0; nb < 8; ++nb)
            for (int j = 0; j < 8; ++j) {
                int m   = row_base + j + half * 8;
                float v = acc[nb][j];
                out[(long long)m * 128 + nb * 16 + l16] = v > 0.f ? v : 0.f;
            }
    }
}

// Per-graph sums + counts. One node per wave.
__global__ void pool_kernel(const float* __restrict__ h, const int* __restrict__ batch,
                            float* __restrict__ sums, float* __restrict__ counts, int N) {
    int tid  = blockIdx.x * 256 + threadIdx.x;
    int node = tid >> 5;
    if (node >= N) return;
    int l    = tid & 31;
    int g    = batch[node];
    float4 v = *(const float4*)(h + (long long)node * 128 + l * 4);
    float* s = sums + g * 128 + l * 4;
    atomicAdd(s + 0, v.x);
    atomicAdd(s + 1, v.y);
    atomicAdd(s + 2, v.z);
    atomicAdd(s + 3, v.w);
    if (l == 0) atomicAdd(counts + g, 1.0f);
}

// out[g][o] = bfc[o] + (sums[g]/max(cnt,1)) · Wfc[o]   (64x16 outputs, tiny)
__global__ void final_kernel(const float* __restrict__ sums, const float* __restrict__ counts,
                             const float* __restrict__ Wfc, const float* __restrict__ bfc,
                             float* __restrict__ out) {
    int tid = blockIdx.x * blockDim.x + threadIdx.x;
    if (tid >= 64 * 16) return;
    int g = tid >> 4, o = tid & 15;
    float c   = counts[g];
    float inv = 1.0f / (c < 1.f ? 1.f : c);
    float a   = bfc[o];
    const float* srow = sums + g * 128;
    const float* wrow = Wfc + o * 128;
    for (int k = 0; k < 128; ++k) a += srow[k] * inv * wrow[k];
    out[tid] = a;
}

extern "C" void kernel_launch(void* const* d_in, const int* in_sizes, int n_in,
                              void* d_out, int out_size, void* d_ws, size_t ws_size,
                              hipStream_t stream) {
    const float* x       = (const float*)d_in[0];
    const int*   ei      = (const int*)d_in[1];
    const int*   batch   = (const int*)d_in[2];
    const float* W1_root = (const float*)d_in[3];
    const float* W1_rel  = (const float*)d_in[4];
    const float* b1      = (const float*)d_in[5];
    const float* W2_root = (const float*)d_in[6];
    const float* W2_rel  = (const float*)d_in[7];
    const float* b2      = (const float*)d_in[8];
    const float* Wfc     = (const float*)d_in[9];
    const float* bfc     = (const float*)d_in[10];
    float*       out     = (float*)d_out;

    const int N = in_sizes[0] / 128;
    const int E = in_sizes[1] / 2;
    const int* src = ei;
    const int* dst = ei + E;

    float* bufA   = (float*)d_ws;                 // agg / h2   : N*128 f32
    float* bufB   = bufA + (size_t)N * 128;       // h1         : N*128 f32
    float* sums   = bufB + (size_t)N * 128;       // 64*128
    float* counts = sums + 64 * 128;              // 64

    const int nAgg       = N * 128;
    const int zeroBlocks = (nAgg / 4 + 255) / 256;
    const int scatBlocks = (int)(((long long)E * 32 + 255) / 256);
    const int gemmBlocks = (N + 127) / 128;
    const int poolBlocks = (int)(((long long)N * 32 + 255) / 256);

    // Layer 1
    zero_kernel<<<zeroBlocks, 256, 0, stream>>>(bufA, nAgg);
    scatter_add<<<scatBlocks, 256, 0, stream>>>(x, src, dst, bufA, E);
    gemm2_bias_relu<<<gemmBlocks, 256, 0, stream>>>(bufA, x, W1_rel, W1_root, b1, bufB, N);

    // Layer 2 (in-place output over agg buffer: each wave writes only rows it read)
    zero_kernel<<<zeroBlocks, 256, 0, stream>>>(bufA, nAgg);
    scatter_add<<<scatBlocks, 256, 0, stream>>>(bufB, src, dst, bufA, E);
    gemm2_bias_relu<<<gemmBlocks, 256, 0, stream>>>(bufA, bufB, W2_rel, W2_root, b2, bufA, N);

    // Pool + FC
    zero_kernel<<<((64 * 128 + 64) / 4 + 255) / 256, 256, 0, stream>>>(sums, 64 * 128 + 64);
    pool_kernel<<<poolBlocks, 256, 0, stream>>>(bufA, batch, sums, counts, N);
    final_kernel<<<4, 256, 0, stream>>>(sums, counts, Wfc, bfc, out);
}